// LSTM_60902636257638
// MI455X (gfx1250) — compile-verified
//
#include <hip/hip_runtime.h>
#include <math.h>

// Problem constants (match reference)
#define T    96     // SEQ_LEN
#define HID  64     // HIDDEN
#define GATES 256   // 4*HIDDEN
#define IN   5      // INPUT_SIZE
#define OUTN 96     // OUT

typedef __attribute__((ext_vector_type(2))) float v2f;
typedef __attribute__((ext_vector_type(8))) float v8f;

__device__ __forceinline__ float sigmoid_(float x) {
    return 1.0f / (1.0f + __expf(-x));
}

// -----------------------------------------------------------------------------
// Kernel 1: single workgroup (256 threads = 8 wave32).
//  Phase A: stage W_hh (stride-65 padded, bank-conflict-free), x, h0 in LDS; c0 in registers.
//  Phase B: 96 sequential LSTM steps; thread g owns gate row g (64 FMA/step,
//           s_h[k] is an LDS broadcast read, W_hh row read is conflict-free).
//  Phase C: linear1 (Hseq 96x64 @ W1^T 64x96) via V_WMMA_F32_16X16X4_F32,
//           36 tiles of 16x16 across 8 waves, K-chain of 16 WMMAs per tile.
// -----------------------------------------------------------------------------
__global__ __launch_bounds__(256) void lstm_fused_kernel(
    const float* __restrict__ x,    // (T, IN)
    const float* __restrict__ h0,   // (HID)
    const float* __restrict__ c0,   // (HID)
    const float* __restrict__ Wih,  // (GATES, IN)
    const float* __restrict__ Whh,  // (GATES, HID)
    const float* __restrict__ bih,  // (GATES)
    const float* __restrict__ bhh,  // (GATES)
    const float* __restrict__ W1,   // (OUTN, HID)
    const float* __restrict__ b1,   // (OUTN)
    float* __restrict__ H1)         // (T, OUTN) workspace
{
    __shared__ float s_whh [GATES * 65];  // 66,560 B, stride 65 -> no bank conflicts
    __shared__ float s_hseq[T * 65];      // 24,960 B, hseq[t][k] at t*65+k
    __shared__ float s_x   [T * IN];      //  1,920 B
    __shared__ float s_g   [GATES];       //  1,024 B
    __shared__ float s_h   [HID];

    const int tid = threadIdx.x;

    // ---- Phase A: stage operands -------------------------------------------
    for (int idx = tid; idx < GATES * HID; idx += 256) {
        int g = idx >> 6, k = idx & 63;
        s_whh[g * 65 + k] = Whh[idx];          // coalesced global read
    }
    for (int idx = tid; idx < T * IN; idx += 256) s_x[idx] = x[idx];
    if (tid < HID) s_h[tid] = h0[tid];

    float wih[IN];
#pragma unroll
    for (int i = 0; i < IN; ++i) wih[i] = Wih[tid * IN + i];
    const float bias = bih[tid] + bhh[tid];
    float creg = (tid < HID) ? c0[tid] : 0.0f;

    __syncthreads();

    // ---- Phase B: sequential scan ------------------------------------------
    const float* wr = &s_whh[tid * 65];
    for (int t = 0; t < T; ++t) {
        const float* xr = &s_x[t * IN];
        float acc = bias;
#pragma unroll
        for (int i = 0; i < IN; ++i) acc += wih[i] * xr[i];

        float a0 = 0.f, a1 = 0.f, a2 = 0.f, a3 = 0.f;
#pragma unroll
        for (int k = 0; k < HID; k += 4) {
            a0 += s_h[k + 0] * wr[k + 0];
            a1 += s_h[k + 1] * wr[k + 1];
            a2 += s_h[k + 2] * wr[k + 2];
            a3 += s_h[k + 3] * wr[k + 3];
        }
        s_g[tid] = acc + ((a0 + a1) + (a2 + a3));
        __syncthreads();

        if (tid < HID) {
            float gi = sigmoid_(s_g[tid]);
            float gf = sigmoid_(s_g[HID + tid]);
            float gg = tanhf  (s_g[2 * HID + tid]);
            float go = sigmoid_(s_g[3 * HID + tid]);
            creg = gf * creg + gi * gg;
            float hn = go * tanhf(creg);
            s_h[tid] = hn;
            s_hseq[t * 65 + tid] = hn;
        }
        __syncthreads();
    }

    // ---- Phase C: linear1 via f32 WMMA -------------------------------------
    // D(96x96) = Hseq(96x64) x W1^T(64x96) + b1, tiled 6x6 in 16x16 tiles.
    const int lane = tid & 31;
    const int wave = tid >> 5;
    const int half = lane >> 4;   // 0 | 1  (K sub-pair select, ISA 16x4 A layout)
    const int l16  = lane & 15;

    for (int tt = wave; tt < 36; tt += 8) {     // wave-uniform: EXEC all 1s
        const int tm = tt / 6, tn = tt % 6;
        const int M  = tm * 16 + l16;           // A row (time index)
        const int N  = tn * 16 + l16;           // B col (output index)
        const int kb = 2 * half;

        v8f acc = {};
#pragma unroll
        for (int k0 = 0; k0 < HID; k0 += 4) {
            v2f a, b;
            a.x = s_hseq[M * 65 + k0 + kb];         // A[M][k0+kb+0]
            a.y = s_hseq[M * 65 + k0 + kb + 1];     // A[M][k0+kb+1]
            b.x = W1[N * HID + k0 + kb];            // W1^T[k][N] == W1[N][k]
            b.y = W1[N * HID + k0 + kb + 1];
            acc = __builtin_amdgcn_wmma_f32_16x16x4_f32(
                /*neg_a=*/false, a, /*neg_b=*/false, b,
                /*c_mod=*/(short)0, acc, /*reuse_a=*/false, /*reuse_b=*/false);
        }

        const float bb = b1[N];
#pragma unroll
        for (int r = 0; r < 8; ++r) {
            int row = tm * 16 + r + 8 * half;   // ISA C/D layout: M = r + 8*(lane/16)
            H1[row * OUTN + N] = acc[r] + bb;
        }
    }
}

// -----------------------------------------------------------------------------
// Kernel 2: linear2. out[o] = b2[o] + dot(H1.flat(9216), W2[o]).
// 96 blocks x 256 threads; float4 loads, LDS tree reduction. ~3.5MB total read
// -> bandwidth-trivial at 23.3 TB/s.
// -----------------------------------------------------------------------------
__global__ __launch_bounds__(256) void linear2_kernel(
    const float* __restrict__ H1,   // (T*OUTN) flat
    const float* __restrict__ W2,   // (OUTN, T*OUTN)
    const float* __restrict__ b2,   // (OUTN)
    float* __restrict__ out)        // (OUTN)
{
    __shared__ float red[256];
    const int o   = blockIdx.x;
    const int tid = threadIdx.x;

    const float4* h4 = (const float4*)H1;
    const float4* w4 = (const float4*)(W2 + (size_t)o * (T * OUTN));

    // Hint the DMA path: prefetch the back half of this W2 row (global_prefetch_b8).
    __builtin_prefetch(w4 + 1152 + tid, 0, 1);

    float s = 0.0f;
#pragma unroll
    for (int k = 0; k < 9; ++k) {               // 9 * 256 * 4 = 9216 elements
        int j = tid + k * 256;
        float4 a = h4[j];
        float4 b = w4[j];
        s += a.x * b.x + a.y * b.y + a.z * b.z + a.w * b.w;
    }

    red[tid] = s;
    __syncthreads();
    for (int off = 128; off > 0; off >>= 1) {
        if (tid < off) red[tid] += red[tid + off];
        __syncthreads();
    }
    if (tid == 0) out[o] = red[0] + b2[o];
}

// -----------------------------------------------------------------------------
extern "C" void kernel_launch(void* const* d_in, const int* in_sizes, int n_in,
                              void* d_out, int out_size, void* d_ws, size_t ws_size,
                              hipStream_t stream)
{
    const float* x   = (const float*)d_in[0];   // input_seq (1,96,5)
    const float* h0  = (const float*)d_in[1];   // h_0 (1,1,64)
    const float* c0  = (const float*)d_in[2];   // c_0 (1,1,64)
    const float* Wih = (const float*)d_in[3];   // (256,5)
    const float* Whh = (const float*)d_in[4];   // (256,64)
    const float* bih = (const float*)d_in[5];   // (256)
    const float* bhh = (const float*)d_in[6];   // (256)
    const float* W1  = (const float*)d_in[7];   // (96,64)
    const float* b1  = (const float*)d_in[8];   // (96)
    const float* W2  = (const float*)d_in[9];   // (96,9216)
    const float* b2  = (const float*)d_in[10];  // (96)

    float* H1 = (float*)d_ws;                   // (96,96) intermediate

    lstm_fused_kernel<<<1, 256, 0, stream>>>(x, h0, c0, Wih, Whh, bih, bhh,
                                             W1, b1, H1);
    linear2_kernel<<<OUTN, 256, 0, stream>>>(H1, W2, b2, (float*)d_out);
}